// GAEBase_26456998543657
// MI455X (gfx1250) — compile-verified
//
#include <hip/hip_runtime.h>

// GCN autoencoder (GAE) for MI455X / gfx1250.
// Per layer: H = h*W (fp32 WMMA 16x16x4, weights staged into LDS via
// global_load_async_to_lds_b32), agg = H*self_norm + b (fused epilogue),
// then edge scatter: agg[dst] += H[src] * edge_norm (global_atomic_add_f32).
// ReLU is fused into the next layer's A-matrix load.

typedef __attribute__((ext_vector_type(2))) float v2f;
typedef __attribute__((ext_vector_type(8))) float v8f;

// Raw 32-bit LDS byte offset of a generic pointer into a __shared__ object.
// (AS3 pointers are 32-bit on gfx1250; go through a 64-bit integer so the
// host semantic pass also accepts the cast.)
template <typename T>
static __device__ __forceinline__ unsigned lds_off_u32(T* p) {
  return (unsigned)(unsigned long long)(T __attribute__((address_space(3)))*)p;
}

// ---------------------------------------------------------------- utilities

__global__ void zero_f32(float* __restrict__ p, int n) {
  int i = blockIdx.x * blockDim.x + threadIdx.x;
  if (i < n) p[i] = 0.0f;
}

// deg[dst] += 1 over incoming edges
__global__ void deg_kernel(const int* __restrict__ dst, float* __restrict__ deg, int E) {
  int e = blockIdx.x * blockDim.x + threadIdx.x;
  if (e < E) {
    __hip_atomic_fetch_add(&deg[dst[e]], 1.0f, __ATOMIC_RELAXED, __HIP_MEMORY_SCOPE_AGENT);
  }
}

// dis[i] = (1 + deg[i])^{-1/2}
__global__ void dis_kernel(const float* __restrict__ deg, float* __restrict__ dis, int N) {
  int i = blockIdx.x * blockDim.x + threadIdx.x;
  if (i < N) dis[i] = rsqrtf(1.0f + deg[i]);
}

// en[e] = dis[src[e]] * dis[dst[e]]  (reused by all 4 layers)
__global__ void edgenorm_kernel(const int* __restrict__ src, const int* __restrict__ dst,
                                const float* __restrict__ dis, float* __restrict__ en, int E) {
  int e = blockIdx.x * blockDim.x + threadIdx.x;
  if (e < E) en[e] = dis[src[e]] * dis[dst[e]];
}

// ---------------------------------------------------------------- WMMA GEMM
// One wave computes a 16-row tile across all NOUT columns using
// V_WMMA_F32_16X16X4_F32 (full fp32 precision).
//   A (16x4 f32): lane l<16 -> row l, {K=k0, k0+1}; lane l>=16 -> row l-16, {k0+2, k0+3}
//   B (4x16 f32): lane l<16 -> col l, rows {k0, k0+1}; lane l>=16 -> col l-16, rows {k0+2, k0+3}
//   C/D (16x16) : VGPR r -> (M=r, N=lane) for lanes 0-15, (M=r+8, N=lane-16) for lanes 16-31
// Weights are staged transposed into LDS (Wt[col][k], row stride K+4 for
// bank-conflict-free b64 reads) with the gfx1250 async copy path.
// Epilogue writes Hw and agg = Hw*dis[row]^2 + bias[col].
template <int K, int NOUT, bool RELU_IN, int WAVES>
__global__ void gcn_gemm(const float* __restrict__ in, const float* __restrict__ W,
                         const float* __restrict__ bias, const float* __restrict__ dis,
                         float* __restrict__ Hw, float* __restrict__ agg, int nTiles) {
  constexpr int KP = K + 4;                     // padded LDS row stride (floats)
  __shared__ float Wt[NOUT * KP];

  // ---- stage W (K x NOUT) transposed into LDS via async global->LDS copies
  {
    const int tid = threadIdx.x;
    for (int idx = tid; idx < K * NOUT; idx += WAVES * 32) {
      const int k = idx / NOUT;
      const int c = idx - k * NOUT;
      const unsigned ldsAddr = lds_off_u32(&Wt[c * KP + k]);
      const unsigned long long ga = (unsigned long long)(W + idx);
      asm volatile("global_load_async_to_lds_b32 %0, %1, off"
                   :: "v"(ldsAddr), "v"(ga) : "memory");
    }
    asm volatile("s_wait_asynccnt 0x0" ::: "memory");
    __syncthreads();
  }

  const int lane = threadIdx.x & 31;
  const int wave = threadIdx.x >> 5;
  const int tile = blockIdx.x * WAVES + wave;   // wave-uniform
  if (tile >= nTiles) return;                   // after barrier; EXEC all-ones past here

  const int rowBase = tile * 16;
  const int half    = lane >> 4;                // 0 or 1
  const int l16     = lane & 15;
  constexpr int NT  = NOUT / 16;

  v8f acc[NT] = {};                             // zero accumulators

  const int aRow = rowBase + l16;
  const float* __restrict__ aPtr = in + (size_t)aRow * K;
  const float* __restrict__ bPtr = &Wt[l16 * KP];  // column l16 of W, +t*16 rows per tile

#pragma unroll 4
  for (int k0 = 0; k0 < K; k0 += 4) {
    const int kk = k0 + half * 2;
    float a0 = aPtr[kk];
    float a1 = aPtr[kk + 1];
    if (RELU_IN) { a0 = fmaxf(a0, 0.0f); a1 = fmaxf(a1, 0.0f); }
    v2f a; a.x = a0; a.y = a1;
#pragma unroll
    for (int t = 0; t < NT; ++t) {
      // Wt[(t*16+l16)*KP + kk], kk even -> single 8-byte LDS load
      const v2f b = *(const v2f*)(bPtr + t * 16 * KP + kk);
      acc[t] = __builtin_amdgcn_wmma_f32_16x16x4_f32(
          /*neg_a=*/false, a, /*neg_b=*/false, b,
          /*c_mod=*/(short)0, acc[t], /*reuse_a=*/false, /*reuse_b=*/false);
    }
  }

  const int rowOff = half * 8;
#pragma unroll
  for (int r = 0; r < 8; ++r) {
    const int row = rowBase + rowOff + r;
    const float d = dis[row];
    const float sn = d * d;                     // self-loop norm = deg^{-1}
    float* __restrict__ hRow = Hw  + (size_t)row * NOUT;
    float* __restrict__ gRow = agg + (size_t)row * NOUT;
#pragma unroll
    for (int t = 0; t < NT; ++t) {
      const int col = t * 16 + l16;
      const float v = acc[t][r];
      hRow[col] = v;
      gRow[col] = fmaf(v, sn, bias[col]);
    }
  }
}

// ---------------------------------------------------------------- scatter-add
// agg[dst[e], :] += Hw[src[e], :] * en[e]; one thread per (edge, float4 chunk)
template <int F>
__global__ void gcn_scatter(const float* __restrict__ Hw, const int* __restrict__ src,
                            const int* __restrict__ dst, const float* __restrict__ en,
                            float* __restrict__ agg, int nEdges) {
  constexpr int C = F / 4;
  const int idx = blockIdx.x * blockDim.x + threadIdx.x;
  if (idx >= nEdges * C) return;
  const int e = idx / C;
  const int c = idx - e * C;

  // speculative prefetch of an upcoming source row (global_prefetch_b8)
  const int ep = e + 256;
  if (ep < nEdges) __builtin_prefetch(Hw + (size_t)src[ep] * F, 0, 0);

  const int s = src[e];
  const int d = dst[e];
  const float w = en[e];
  const float4 v = *reinterpret_cast<const float4*>(Hw + (size_t)s * F + c * 4);
  float* __restrict__ p = agg + (size_t)d * F + c * 4;
  __hip_atomic_fetch_add(p + 0, v.x * w, __ATOMIC_RELAXED, __HIP_MEMORY_SCOPE_AGENT);
  __hip_atomic_fetch_add(p + 1, v.y * w, __ATOMIC_RELAXED, __HIP_MEMORY_SCOPE_AGENT);
  __hip_atomic_fetch_add(p + 2, v.z * w, __ATOMIC_RELAXED, __HIP_MEMORY_SCOPE_AGENT);
  __hip_atomic_fetch_add(p + 3, v.w * w, __ATOMIC_RELAXED, __HIP_MEMORY_SCOPE_AGENT);
}

// ---------------------------------------------------------------- host side

extern "C" void kernel_launch(void* const* d_in, const int* in_sizes, int n_in,
                              void* d_out, int out_size, void* d_ws, size_t ws_size,
                              hipStream_t stream) {
  constexpr int IN_DIM  = 128;
  constexpr int HID_DIM = 64;

  const int N = in_sizes[0] / IN_DIM;   // 50000
  const int E = in_sizes[1] / 2;        // 800000

  const float* x  = (const float*)d_in[0];
  // edge_index [2,E]; JAX default x64-disabled -> int32
  const int* src  = (const int*)d_in[1];
  const int* dst  = src + E;
  const float* W1 = (const float*)d_in[2]; const float* b1 = (const float*)d_in[3];
  const float* W2 = (const float*)d_in[4]; const float* b2 = (const float*)d_in[5];
  const float* W3 = (const float*)d_in[6]; const float* b3 = (const float*)d_in[7];
  const float* Wd = (const float*)d_in[8]; const float* bd = (const float*)d_in[9];
  float* out = (float*)d_out;

  // workspace carve-up (fp32): deg[N] dis[N] en[E] B0[N*128] B1[N*64] B2[N*64]
  float* deg = (float*)d_ws;
  float* dis = deg + N;
  float* en  = dis + N;
  float* B0  = en + E;                    // Hw buffer (up to N x 128)
  float* B1  = B0 + (size_t)N * IN_DIM;   // agg / activations (N x 64)
  float* B2  = B1 + (size_t)N * HID_DIM;  // agg / activations (N x 64)

  const int nTiles = (N + 15) / 16;       // 3125 (N divisible by 16)
  constexpr int WAVES = 4;
  const dim3 gemmBlk(32 * WAVES);
  const dim3 gemmGrd((nTiles + WAVES - 1) / WAVES);

  const int TB = 256;
  auto blocks = [&](long long n) { return dim3((unsigned)((n + TB - 1) / TB)); };

  // ---- normalization pre-pass (once; reused by all layers)
  zero_f32<<<blocks(N), TB, 0, stream>>>(deg, N);
  deg_kernel<<<blocks(E), TB, 0, stream>>>(dst, deg, E);
  dis_kernel<<<blocks(N), TB, 0, stream>>>(deg, dis, N);
  edgenorm_kernel<<<blocks(E), TB, 0, stream>>>(src, dst, dis, en, E);

  const long long eh = (long long)E * (HID_DIM / 4);
  const long long ei = (long long)E * (IN_DIM / 4);

  // ---- layer 1: x(128) -> B1(64)       [ReLU consumed by layer 2]
  gcn_gemm<IN_DIM, HID_DIM, false, WAVES>
      <<<gemmGrd, gemmBlk, 0, stream>>>(x, W1, b1, dis, B0, B1, nTiles);
  gcn_scatter<HID_DIM><<<blocks(eh), TB, 0, stream>>>(B0, src, dst, en, B1, E);

  // ---- layer 2: relu(B1)(64) -> B2(64) [ReLU consumed by layer 3]
  gcn_gemm<HID_DIM, HID_DIM, true, WAVES>
      <<<gemmGrd, gemmBlk, 0, stream>>>(B1, W2, b2, dis, B0, B2, nTiles);
  gcn_scatter<HID_DIM><<<blocks(eh), TB, 0, stream>>>(B0, src, dst, en, B2, E);

  // ---- layer 3: relu(B2)(64) -> B1(64) = emb (no activation after)
  gcn_gemm<HID_DIM, HID_DIM, true, WAVES>
      <<<gemmGrd, gemmBlk, 0, stream>>>(B2, W3, b3, dis, B0, B1, nTiles);
  gcn_scatter<HID_DIM><<<blocks(eh), TB, 0, stream>>>(B0, src, dst, en, B1, E);

  // ---- decoder: B1(64) -> out(128); agg buffer IS d_out (fully initialized by epilogue)
  gcn_gemm<HID_DIM, IN_DIM, false, WAVES>
      <<<gemmGrd, gemmBlk, 0, stream>>>(B1, Wd, bd, dis, B0, out, nTiles);
  gcn_scatter<IN_DIM><<<blocks(ei), TB, 0, stream>>>(B0, src, dst, en, out, E);
}